// corrRISE_83975200571792
// MI455X (gfx1250) — compile-verified
//
#include <hip/hip_runtime.h>
#include <math.h>

// ---------------- problem constants ----------------
#define HWPIX   50176              // 224*224
#define CH      3
#define NMASK   1000
#define EMB     512
#define KDIM    HWPIX              // GEMM K
#define MPAD    1024               // padded mask count
#define NEFF    1024               // B1 (0..511) ++ B2 (512..1023)

// ---------------- GEMM tiling ----------------------
#define BM      64
#define BN      128
#define KC      16
#define ASTR    (KC + 4)           // 20: rows 16B-aligned, 5m mod 16 distinct -> no bank conflicts
#define BSTR    (BN + 8)           // 136: row-pair bank offset 16 -> disjoint fragment banks
#define KSPLIT  4
#define KLEN    (KDIM / KSPLIT)    // 12544, divisible by KC

typedef __attribute__((ext_vector_type(2))) float v2f;
typedef __attribute__((ext_vector_type(8))) float v8f;

// =====================================================================
// init: zero the atomic-accumulated regions (XM, xy) each call
// =====================================================================
__global__ void k_init(float* __restrict__ p, int n) {
    int i = blockIdx.x * blockDim.x + threadIdx.x;
    if (i < n) p[i] = 0.0f;
}

// =====================================================================
// B1[pix,d] = sum_c img1[c,pix] * W[(c*HW+pix)*512 + d]   (and B2)
// =====================================================================
__global__ __launch_bounds__(256)
void k_buildB(const float* __restrict__ img1, const float* __restrict__ img2,
              const float* __restrict__ W,
              float* __restrict__ B1, float* __restrict__ B2) {
    size_t gid = (size_t)blockIdx.x * 256 + threadIdx.x;   // pix*512 + d
    int pix = (int)(gid >> 9);
    int d   = (int)(gid & 511);
    float s1 = 0.f, s2 = 0.f;
#pragma unroll
    for (int c = 0; c < CH; ++c) {
        size_t p = (size_t)(c * HWPIX + pix);
        float w = W[p * EMB + d];
        s1 = fmaf(img1[p], w, s1);
        s2 = fmaf(img2[p], w, s2);
    }
    B1[gid] = s1;
    B2[gid] = s2;
}

// =====================================================================
// full-image embeddings: x[d] = colsum(B1), y[d] = colsum(B2)
// =====================================================================
__global__ __launch_bounds__(256)
void k_colsum(const float* __restrict__ B1, const float* __restrict__ B2,
              float* __restrict__ xy) {
    int d = blockIdx.x * 256 + threadIdx.x;                // 0..511
    const float* __restrict__ B = blockIdx.z ? B2 : B1;
    size_t p0 = (size_t)blockIdx.y * 1024;
    float acc = 0.f;
    for (int p = 0; p < 1024; ++p)
        acc += B[(p0 + p) * EMB + d];
    atomicAdd(&xy[blockIdx.z * EMB + d], acc);
}

// =====================================================================
// main GEMM:  XM[m, 0..1023] += masks[m, kseg] @ [B1|B2][kseg, :]
// fp32 WMMA 16x16x4; double-buffered LDS tiles filled by
// global_load_async_to_lds_b128 (ASYNCcnt), overlapped with compute.
// 8 waves per WG arranged 2(M) x 4(N); each wave owns a 32x32 patch.
// =====================================================================
__global__ __launch_bounds__(256)
void k_gemm(const float* __restrict__ masks, const float* __restrict__ B1,
            const float* __restrict__ B2, float* __restrict__ XM) {
    __shared__ float As[2][BM][ASTR];
    __shared__ float Bs[2][KC][BSTR];

    const int tid  = threadIdx.x;
    const int wave = tid >> 5;
    const int lane = tid & 31;
    const int hi   = lane >> 4;     // half-wave select (K offset +2, row +8)
    const int ln   = lane & 15;

    const int nb = blockIdx.x * BN;        // 0..896   (>=512 -> B2 columns)
    const int mb = blockIdx.y * BM;        // 0..960
    const int kb = blockIdx.z * KLEN;

    const int wm = wave >> 2;              // 0..1
    const int wn = wave & 3;               // 0..3
    const int m0 = wm * 32;
    const int n0 = wn * 32;

    const float* __restrict__ bsrc = (nb < EMB) ? (B1 + nb) : (B2 + (nb - EMB));

    v8f c00 = {}, c01 = {}, c10 = {}, c11 = {};

    // A-tile coords: 64 rows x 4 float4 (16B per thread per tile)
    const int arow = tid >> 2;
    const int aq   = (tid & 3) * 4;
    const bool arow_ok = (mb + arow) < NMASK;
    const float* __restrict__ aptr =
        masks + (size_t)(mb + arow) * KDIM + kb + aq;
    // B-tile coords: 16 rows x 2 float4 per thread
    const int bkr = tid >> 4;
    const int bq  = (tid & 15) * 4;
    const float* __restrict__ bptr = bsrc + (size_t)(kb + bkr) * EMB;

    // issue async copies of tile at K-offset kt into buffer `buf`
    auto issue = [&](int buf, int kt) {
        if (arow_ok) {
            const float* ga = aptr + kt;
            unsigned dstA = (unsigned)(uintptr_t)&As[buf][arow][aq];
            asm volatile("global_load_async_to_lds_b128 %0, %1, off"
                         :: "v"(dstA), "v"(ga) : "memory");
        }
        const float* gb0 = bptr + (size_t)kt * EMB + bq;
        unsigned dstB0 = (unsigned)(uintptr_t)&Bs[buf][bkr][bq];
        asm volatile("global_load_async_to_lds_b128 %0, %1, off"
                     :: "v"(dstB0), "v"(gb0) : "memory");
        const float* gb1 = gb0 + 64;
        unsigned dstB1 = (unsigned)(uintptr_t)&Bs[buf][bkr][bq + 64];
        asm volatile("global_load_async_to_lds_b128 %0, %1, off"
                     :: "v"(dstB1), "v"(gb1) : "memory");
        // warm L2 one tile further ahead
        if (kt + KC < KLEN) {
            if (arow_ok) __builtin_prefetch(aptr + kt + KC, 0, 1);
            __builtin_prefetch(bptr + (size_t)(kt + KC) * EMB + bq, 0, 1);
        }
    };

    // prologue: zero-pad OOB mask rows once (never async-overwritten)
    if (!arow_ok) {
#pragma unroll
        for (int j = 0; j < 4; ++j) {
            As[0][arow][aq + j] = 0.f;
            As[1][arow][aq + j] = 0.f;
        }
    }
    issue(0, 0);
    asm volatile("s_wait_asynccnt 0x0" ::: "memory");
    __syncthreads();

    for (int kt = 0; kt < KLEN; kt += KC) {
        const int buf = (kt / KC) & 1;
        if (kt + KC < KLEN) issue(buf ^ 1, kt + KC);   // fill next while computing
#pragma unroll
        for (int kk = 0; kk < KC; kk += 4) {
            const int ka = kk + 2 * hi;    // VGPR j <-> K = kk + j + 2*hi
            v2f a0 = { As[buf][m0 + ln][ka],      As[buf][m0 + ln][ka + 1] };
            v2f a1 = { As[buf][m0 + 16 + ln][ka], As[buf][m0 + 16 + ln][ka + 1] };
            v2f b0 = { Bs[buf][ka][n0 + ln],      Bs[buf][ka + 1][n0 + ln] };
            v2f b1 = { Bs[buf][ka][n0 + 16 + ln], Bs[buf][ka + 1][n0 + 16 + ln] };
            c00 = __builtin_amdgcn_wmma_f32_16x16x4_f32(false, a0, false, b0,
                                                        (short)0, c00, false, false);
            c01 = __builtin_amdgcn_wmma_f32_16x16x4_f32(false, a0, false, b1,
                                                        (short)0, c01, false, false);
            c10 = __builtin_amdgcn_wmma_f32_16x16x4_f32(false, a1, false, b0,
                                                        (short)0, c10, false, false);
            c11 = __builtin_amdgcn_wmma_f32_16x16x4_f32(false, a1, false, b1,
                                                        (short)0, c11, false, false);
        }
        // my async writes into buf^1 done; barrier makes them (and everyone's
        // reads of buf) visible before the next iteration flips buffers
        asm volatile("s_wait_asynccnt 0x0" ::: "memory");
        __syncthreads();
    }

    // C/D layout: VGPR r <-> row M = r + 8*hi, col N = lane&15
    const int col0 = nb + n0 + ln;
    const int col1 = nb + n0 + 16 + ln;
#pragma unroll
    for (int r = 0; r < 8; ++r) {
        const int row0 = mb + m0 + r + 8 * hi;
        const int row1 = row0 + 16;
        if (row0 < NMASK) {
            atomicAdd(&XM[(size_t)row0 * NEFF + col0], c00[r]);
            atomicAdd(&XM[(size_t)row0 * NEFF + col1], c01[r]);
        }
        if (row1 < NMASK) {
            atomicAdd(&XM[(size_t)row1 * NEFF + col0], c10[r]);
            atomicAdd(&XM[(size_t)row1 * NEFF + col1], c11[r]);
        }
    }
}

// =====================================================================
// cosine: sc_x[n] = (x_m[n]·y)/(|x_m[n]||y|),  sc_y[n] = (y_m[n]·x)/(|y_m[n]||x|)
// =====================================================================
__global__ __launch_bounds__(256)
void k_cosine(const float* __restrict__ XM, const float* __restrict__ xy,
              float* __restrict__ sc) {
    const int n = blockIdx.x;
    const int t = threadIdx.x;
    float dxy = 0, nx = 0, dyx = 0, ny = 0, yy = 0, xx = 0;
    for (int d = t; d < EMB; d += 256) {
        float xv = xy[d], yv = xy[EMB + d];
        float xm = XM[(size_t)n * NEFF + d];
        float ym = XM[(size_t)n * NEFF + EMB + d];
        dxy = fmaf(xm, yv, dxy); nx = fmaf(xm, xm, nx);
        dyx = fmaf(ym, xv, dyx); ny = fmaf(ym, ym, ny);
        yy  = fmaf(yv, yv, yy);  xx = fmaf(xv, xv, xx);
    }
    __shared__ float red[6][256];
    red[0][t] = dxy; red[1][t] = nx; red[2][t] = dyx;
    red[3][t] = ny;  red[4][t] = yy; red[5][t] = xx;
    __syncthreads();
    for (int s = 128; s > 0; s >>= 1) {
        if (t < s)
            for (int j = 0; j < 6; ++j) red[j][t] += red[j][t + s];
        __syncthreads();
    }
    if (t == 0) {
        sc[n]        = red[0][0] / (sqrtf(red[1][0]) * sqrtf(red[4][0]));
        sc[1024 + n] = red[2][0] / (sqrtf(red[3][0]) * sqrtf(red[5][0]));
    }
}

// =====================================================================
// center scores, produce sc_c arrays and their L2 norms
// =====================================================================
__global__ __launch_bounds__(1024)
void k_center(const float* __restrict__ sc, float* __restrict__ scc,
              float* __restrict__ stats) {
    const int t = threadIdx.x;
    __shared__ float red[1024];
    for (int half = 0; half < 2; ++half) {
        float v = (t < NMASK) ? sc[half * 1024 + t] : 0.f;
        red[t] = v; __syncthreads();
        for (int s = 512; s > 0; s >>= 1) {
            if (t < s) red[t] += red[t + s];
            __syncthreads();
        }
        float mean = red[0] * (1.0f / NMASK);
        __syncthreads();
        float c = (t < NMASK) ? (v - mean) : 0.f;
        scc[half * 1024 + t] = c;
        red[t] = c * c; __syncthreads();
        for (int s = 512; s > 0; s >>= 1) {
            if (t < s) red[t] += red[t + s];
            __syncthreads();
        }
        if (t == 0) stats[half] = sqrtf(red[0]);
        __syncthreads();
    }
}

// =====================================================================
// per-pixel Pearson: num = sc_c · masks[:,pix] (mask centering cancels),
// den = |sc_c| * sqrt(Σm² - N·mean²)
// =====================================================================
__global__ __launch_bounds__(256)
void k_pearson(const float* __restrict__ masks, const float* __restrict__ scc,
               const float* __restrict__ stats, float* __restrict__ out) {
    __shared__ float sx[NMASK], sy[NMASK];
    const int t = threadIdx.x;
    for (int i = t; i < NMASK; i += 256) { sx[i] = scc[i]; sy[i] = scc[1024 + i]; }
    __syncthreads();
    const int pix = blockIdx.x * 256 + t;
    float s = 0, ss = 0, dx = 0, dy = 0;
    for (int n = 0; n < NMASK; ++n) {
        float m = masks[(size_t)n * HWPIX + pix];
        s  += m;
        ss  = fmaf(m, m, ss);
        dx  = fmaf(sx[n], m, dx);
        dy  = fmaf(sy[n], m, dy);
    }
    float mean = s * (1.0f / NMASK);
    float varm = ss - (float)NMASK * mean * mean;
    float denb = sqrtf(varm);
    out[pix]          = dx / (stats[0] * denb);
    out[HWPIX + pix]  = dy / (stats[1] * denb);
}

// =====================================================================
// launcher
// =====================================================================
extern "C" void kernel_launch(void* const* d_in, const int* in_sizes, int n_in,
                              void* d_out, int out_size, void* d_ws, size_t ws_size,
                              hipStream_t stream) {
    const float* img1  = (const float*)d_in[0];   // [1,3,224,224]
    const float* img2  = (const float*)d_in[1];
    const float* masks = (const float*)d_in[2];   // [1000,1,224,224]
    const float* W     = (const float*)d_in[3];   // [150528,512]
    float* out = (float*)d_out;                   // [2,224,224]

    // workspace layout (floats); ~210 MB total
    float* B1    = (float*)d_ws;
    float* B2    = B1 + (size_t)HWPIX * EMB;
    float* XM    = B2 + (size_t)HWPIX * EMB;        // [1024,1024]
    float* xy    = XM + (size_t)MPAD * NEFF;        // x:0..511, y:512..1023
    float* sc    = xy + 1024;                       // raw scores (2x1024)
    float* scc   = sc + 2048;                       // centered   (2x1024)
    float* stats = scc + 2048;                      // [0]=|scx_c|, [1]=|scy_c|

    // 1) zero atomic accumulators (XM + xy)
    {
        int n = MPAD * NEFF + 1024;
        k_init<<<(n + 255) / 256, 256, 0, stream>>>(XM, n);
    }
    // 2) fold images into projection matrix
    k_buildB<<<(HWPIX * EMB) / 256, 256, 0, stream>>>(img1, img2, W, B1, B2);
    // 3) full-image embeddings = column sums
    k_colsum<<<dim3(2, HWPIX / 1024, 2), 256, 0, stream>>>(B1, B2, xy);
    // 4) fp32 WMMA GEMM, async-LDS double buffered, split-K x4
    k_gemm<<<dim3(NEFF / BN, MPAD / BM, KSPLIT), 256, 0, stream>>>(masks, B1, B2, XM);
    // 5) cosine similarities
    k_cosine<<<NMASK, 256, 0, stream>>>(XM, xy, sc);
    // 6) center scores
    k_center<<<1, 1024, 0, stream>>>(sc, scc, stats);
    // 7) per-pixel Pearson -> output saliency maps
    k_pearson<<<HWPIX / 256, 256, 0, stream>>>(masks, scc, stats, out);

    (void)in_sizes; (void)n_in; (void)out_size; (void)ws_size;
}